// VecArrayMultiplier54x54_43860206027325
// MI455X (gfx1250) — compile-verified
//
#include <hip/hip_runtime.h>
#include <stdint.h>

// 54x54 -> 108-bit boolean array multiplier.
// Inputs are exact {0,1} floats, so the differentiable gate network is bit-exactly
// an integer multiply. Strategy: wave32-per-row.
//   1) async-stage the row (54+54 floats) into LDS via global_load_async_to_lds_b64
//      (CDNA5 async data-mover path, ASYNCcnt-tracked)
//   2) ballot-pack the {0,1} lanes into two u64 operands (wave32 ballot)
//   3) 64x64->128 multiply (wave-uniform -> scalar ALU)
//   4) each of 27 lanes stores 4 product bits as one float4 -> single
//      global_store_b128 per wave, fully coalesced (108 floats = 27 x 16B)
// Memory-bound: ~7.1 MB total @ 23.3 TB/s => ~0.3 us floor; L2-resident (192 MB).

#define BITS 54
#define OUT_BITS 108
#define WAVES_PER_BLOCK 8
#define LDS_STRIDE 128  // floats per wave: [0,54) = A row, [64,118) = B row

static __device__ __forceinline__ uint32_t lds_addr32(const void* p) {
  // generic pointer to __shared__ data: low 32 bits of the flat address are the
  // LDS byte offset, which is what the DS/async hardware consumes.
  return (uint32_t)(uintptr_t)p;
}

static __device__ __forceinline__ uint32_t ballot32(bool pred) {
#if defined(__has_builtin) && __has_builtin(__builtin_amdgcn_ballot_w32)
  return __builtin_amdgcn_ballot_w32(pred);
#else
  return (uint32_t)__ballot(pred);
#endif
}

__global__ __launch_bounds__(256)
void vec_mul54x54_kernel(const float* __restrict__ A,
                         const float* __restrict__ B,
                         float* __restrict__ out,
                         int rows) {
  __shared__ float smem[WAVES_PER_BLOCK * LDS_STRIDE];

  const int lane = threadIdx.x & 31;
  const int wave = threadIdx.x >> 5;
  const int row  = (int)blockIdx.x * WAVES_PER_BLOCK + wave;
  if (row >= rows) return;  // wave-uniform exit

  const float* aRow = A + (size_t)row * BITS;
  const float* bRow = B + (size_t)row * BITS;
  float* sA = &smem[wave * LDS_STRIDE];
  float* sB = sA + 64;

  // --- Stage the row into LDS with async global->LDS loads (27 lanes x 8B). ---
  if (lane < 27) {
    uint64_t ga = (uint64_t)(uintptr_t)(aRow + 2 * lane);
    uint64_t gb = (uint64_t)(uintptr_t)(bRow + 2 * lane);
    uint32_t la = lds_addr32(sA + 2 * lane);
    uint32_t lb = lds_addr32(sB + 2 * lane);
    asm volatile("global_load_async_to_lds_b64 %0, %1, off"
                 :: "v"(la), "v"(ga) : "memory");
    asm volatile("global_load_async_to_lds_b64 %0, %1, off"
                 :: "v"(lb), "v"(gb) : "memory");
  }
  asm volatile("s_wait_asynccnt 0x0" ::: "memory");

  const float a0 = sA[lane];
  const float a1 = (lane < BITS - 32) ? sA[32 + lane] : 0.0f;
  const float b0 = sB[lane];
  const float b1 = (lane < BITS - 32) ? sB[32 + lane] : 0.0f;

  // --- Pack bits across the wave32 lanes: bit i of the mask = lane i's value. ---
  const uint32_t mA0 = ballot32(a0 > 0.5f);
  const uint32_t mA1 = ballot32(a1 > 0.5f);  // lanes >= 22 contributed 0
  const uint32_t mB0 = ballot32(b0 > 0.5f);
  const uint32_t mB1 = ballot32(b1 > 0.5f);

  const uint64_t av = (uint64_t)mA0 | ((uint64_t)mA1 << 32);
  const uint64_t bv = (uint64_t)mB0 | ((uint64_t)mB1 << 32);

  // --- 54b x 54b -> 108b product (wave-uniform -> scalar ALU). ---
  const uint64_t lo = av * bv;            // product bits [63:0]
  const uint64_t hi = __umul64hi(av, bv); // product bits [127:64]

  // --- Unpack: lane L (L<27) writes product bits [4L .. 4L+3] as one float4. ---
  // 4L and 64 are both multiples of 4, so a lane's nibble never straddles lo/hi.
  if (lane < OUT_BITS / 4) {
    const uint32_t sh = 4u * (uint32_t)lane;
    const uint32_t nib =
        (uint32_t)((lane < 16) ? (lo >> sh) : (hi >> (sh - 64))) & 0xFu;
    float4 v;
    v.x = (float)(nib & 1u);
    v.y = (float)((nib >> 1) & 1u);
    v.z = (float)((nib >> 2) & 1u);
    v.w = (float)((nib >> 3) & 1u);
    // out row is 432 B = 27 * 16 B; base and per-lane offsets are 16B-aligned.
    float4* o4 = reinterpret_cast<float4*>(out + (size_t)row * OUT_BITS) + lane;
    *o4 = v;  // -> global_store_b128
  }
}

extern "C" void kernel_launch(void* const* d_in, const int* in_sizes, int n_in,
                              void* d_out, int out_size, void* d_ws, size_t ws_size,
                              hipStream_t stream) {
  const float* A = (const float*)d_in[0];
  const float* B = (const float*)d_in[1];
  float* out = (float*)d_out;

  const int rows = in_sizes[0] / BITS;  // 8192 for the reference setup
  const int blocks = (rows + WAVES_PER_BLOCK - 1) / WAVES_PER_BLOCK;

  hipLaunchKernelGGL(vec_mul54x54_kernel, dim3(blocks), dim3(256), 0, stream,
                     A, B, out, rows);
}